// PyG_SGC_29635274342814
// MI455X (gfx1250) — compile-verified
//
#include <hip/hip_runtime.h>
#include <hip/hip_bf16.h>

typedef __attribute__((ext_vector_type(2))) float v2f;
typedef __attribute__((ext_vector_type(8))) float v8f;

// ---------------- utility kernels ----------------

__global__ __launch_bounds__(256) void zero_f4(float4* p, long long n4) {
    long long i = (long long)blockIdx.x * blockDim.x + threadIdx.x;
    if (i < n4) p[i] = make_float4(0.f, 0.f, 0.f, 0.f);
}

__global__ __launch_bounds__(256) void init_deg(float* deg, int n) {
    int i = blockIdx.x * blockDim.x + threadIdx.x;
    if (i < n) deg[i] = 1.0f;   // self-loop contributes 1 to every node's in-degree
}

__global__ __launch_bounds__(256) void count_deg(const int* __restrict__ dst,
                                                 float* __restrict__ deg, int ne) {
    int e = blockIdx.x * blockDim.x + threadIdx.x;
    if (e < ne) atomicAdd(&deg[dst[e]], 1.0f);
}

__global__ __launch_bounds__(256) void deg_to_dis(float* deg, int n) {
    int i = blockIdx.x * blockDim.x + threadIdx.x;
    if (i < n) {
        float d = deg[i];
        deg[i] = (d > 0.0f) ? rsqrtf(d) : 0.0f;
    }
}

// ---------------- propagation: one wave32 per edge ----------------
// Lane l handles floats [4l, 4l+4) of the 128-wide feature row:
// 32 lanes x float4 = 512B fully-coalesced gather, then 4 float atomics
// (global_atomic_add_f32) into the destination row.

__global__ __launch_bounds__(256) void hop_kernel(const int* __restrict__ src,
                                                  const int* __restrict__ dst,
                                                  const float* __restrict__ dis,
                                                  const float* __restrict__ xin,
                                                  float* __restrict__ yout,
                                                  int ne, int n) {
    long long gid  = (long long)blockIdx.x * blockDim.x + threadIdx.x;
    long long wave = gid >> 5;
    int lane = (int)(gid & 31);
    long long etot = (long long)ne + n;
    if (wave >= etot) return;

    int s, d;
    if (wave < ne) { s = src[wave]; d = dst[wave]; }
    else           { s = d = (int)(wave - ne); }   // self loop

    float nrm = dis[s] * dis[d];

    const float4* xr = (const float4*)(xin + (size_t)s * 128);
    float4 v = xr[lane];

    float* yo = yout + (size_t)d * 128 + (size_t)lane * 4;
    atomicAdd(yo + 0, nrm * v.x);
    atomicAdd(yo + 1, nrm * v.y);
    atomicAdd(yo + 2, nrm * v.z);
    atomicAdd(yo + 3, nrm * v.w);
}

// ---------------- final linear via V_WMMA_F32_16X16X4_F32 ----------------
// One wave computes a 16x16 output tile; 4 waves per block cover all 64
// output columns of a 16-node row group. K = 128 -> 32 WMMA steps.
//
// Layouts per CDNA5 ISA 7.12.2 (wave32):
//  A (16x4 f32):  lane -> M = lane&15 ; VGPR{0,1} hold K = (lane>>4)*2 , +1
//  B (4x16 f32):  lane -> N = lane&15 ; VGPR{0,1} hold K = (lane>>4)*2 , +1
//  C/D (16x16):   VGPR i: lanes 0-15 -> (M=i, N=lane) ; lanes 16-31 -> (M=i+8, N=lane-16)

__global__ __launch_bounds__(128) void gemm_wmma(const float* __restrict__ X2,
                                                 const float* __restrict__ W,   // [64,128] row-major
                                                 const float* __restrict__ bias,
                                                 float* __restrict__ out,      // [n,64]
                                                 int n) {
    int lane = threadIdx.x & 31;
    int wv   = threadIdx.x >> 5;      // 0..3 -> 16-column tile
    int row0 = blockIdx.x * 16;
    int col0 = wv * 16;
    if (row0 >= n) return;

    int mn = lane & 15;               // M for A, N for B/C
    int kb = (lane >> 4) * 2;         // K sub-offset within the 4-wide step

    // C initialized with bias: every row of a column gets bias[col]
    float bv = bias[col0 + mn];
    v8f c;
#pragma unroll
    for (int i = 0; i < 8; ++i) c[i] = bv;

    const float* arow = X2 + (size_t)(row0 + mn) * 128 + kb;  // A[m][k]
    const float* brow = W  + (size_t)(col0 + mn) * 128 + kb;  // B[k][n] = W[n][k]

#pragma unroll 4
    for (int k = 0; k < 128; k += 4) {
        v2f a, b;
        a.x = arow[k];  a.y = arow[k + 1];
        b.x = brow[k];  b.y = brow[k + 1];
        // 8 args: (neg_a, A, neg_b, B, c_mod, C, reuse_a, reuse_b)
        c = __builtin_amdgcn_wmma_f32_16x16x4_f32(false, a, false, b,
                                                  (short)0, c, false, false);
    }

    // store D
    int ncol  = col0 + mn;
    int mbase = row0 + ((lane >> 4) * 8);
#pragma unroll
    for (int i = 0; i < 8; ++i)
        out[(size_t)(mbase + i) * 64 + ncol] = c[i];
}

// ---------------- host-side launcher ----------------

extern "C" void kernel_launch(void* const* d_in, const int* in_sizes, int n_in,
                              void* d_out, int out_size, void* d_ws, size_t ws_size,
                              hipStream_t stream) {
    // setup_inputs order: V(scalar), E(2,NE) int, X(N,128) f32, W(64,128) f32, b(64) f32
    const int*   E = (const int*)d_in[1];
    const float* X = (const float*)d_in[2];
    const float* W = (const float*)d_in[3];
    const float* b = (const float*)d_in[4];

    const int n  = in_sizes[2] / 128;   // 100000
    const int ne = in_sizes[1] / 2;     // 1600000
    const int* src = E;
    const int* dst = E + ne;

    // workspace: [deg/dis: n f32][y1: n*128 f32][y2: n*128 f32]  (~103 MB)
    float* ws   = (float*)d_ws;
    size_t y1o  = ((size_t)n + 127) & ~(size_t)127;
    float* deg  = ws;
    float* y1   = ws + y1o;
    float* y2   = y1 + (size_t)n * 128;

    // zero both hop buffers (contiguous) every call — harness does not re-zero ws
    long long n4 = ((long long)n * 128 * 2) / 4;
    zero_f4<<<(int)((n4 + 255) / 256), 256, 0, stream>>>((float4*)y1, n4);

    init_deg<<<(n + 255) / 256, 256, 0, stream>>>(deg, n);
    count_deg<<<(ne + 255) / 256, 256, 0, stream>>>(dst, deg, ne);
    deg_to_dis<<<(n + 255) / 256, 256, 0, stream>>>(deg, n);

    long long etot = (long long)ne + n;
    long long thr  = etot * 32;                 // one wave32 per edge
    int hop_blocks = (int)((thr + 255) / 256);
    hop_kernel<<<hop_blocks, 256, 0, stream>>>(src, dst, deg, X,  y1, ne, n);
    hop_kernel<<<hop_blocks, 256, 0, stream>>>(src, dst, deg, y1, y2, ne, n);

    gemm_wmma<<<n / 16, 128, 0, stream>>>(y2, W, b, (float*)d_out, n);
}